// HardestPositiveContrastiveLoss_88751204204758
// MI455X (gfx1250) — compile-verified
//
#include <hip/hip_runtime.h>
#include <hip/hip_bf16.h>
#include <math.h>

typedef __attribute__((ext_vector_type(2))) float v2f;
typedef __attribute__((ext_vector_type(8))) float v8f;

namespace {
constexpr int kD      = 256;               // feature dim
constexpr int kPairs  = 4096;              // NUM_PAIRS
constexpr int kTopK   = 100;               // TO_PICK
constexpr int kTile   = 64;                // distance tile edge per block
constexpr int kGrid   = kPairs / kTile;    // 64
constexpr int kBlocks = kGrid * kGrid;     // 4096
constexpr int kKC     = 64;                // K staging chunk in LDS
constexpr int kPitch  = 68;                // LDS row pitch (bank-conflict pad, 16B aligned)
}

// ---------------------------------------------------------------------------
// Kernel 1: gather rows of the feature map into contiguous c1/c2 and compute
// per-row squared norms. One block (256 threads) per pair index.
// ---------------------------------------------------------------------------
__global__ void __launch_bounds__(256)
gather_sq_kernel(const float* __restrict__ fm,
                 const int* __restrict__ ids1,
                 const int* __restrict__ ids2,
                 float* __restrict__ c1, float* __restrict__ c2,
                 float* __restrict__ sq1, float* __restrict__ sq2)
{
  __shared__ float red[256];
  const int b = blockIdx.x;
  const int t = threadIdx.x;

  const long long r1 = (long long)ids1[b];
  const float v1 = fm[r1 * kD + t];
  c1[(size_t)b * kD + t] = v1;
  red[t] = v1 * v1;
  __syncthreads();
  for (int s = 128; s > 0; s >>= 1) {
    if (t < s) red[t] += red[t + s];
    __syncthreads();
  }
  if (t == 0) sq1[b] = red[0];
  __syncthreads();

  const long long r2 = (long long)ids2[b];
  const float v2_ = fm[r2 * kD + t];
  c2[(size_t)b * kD + t] = v2_;
  red[t] = v2_ * v2_;
  __syncthreads();
  for (int s = 128; s > 0; s >>= 1) {
    if (t < s) red[t] += red[t + s];
    __syncthreads();
  }
  if (t == 0) sq2[b] = red[0];
}

// ---------------------------------------------------------------------------
// Kernel 2: one 64x64 distance tile per block. 8 wave32s; each wave owns a
// 32(row) x 16(col) strip = two 16x16 WMMA accumulators. K is staged through
// LDS in 64-wide chunks. fp32 WMMA (16x16x4) keeps exact reference precision.
// Then per-tile top-100 extraction into the candidate buffer.
// ---------------------------------------------------------------------------
__global__ void __launch_bounds__(256)
dist_topk_kernel(const float* __restrict__ c1, const float* __restrict__ c2,
                 const float* __restrict__ sq1, const float* __restrict__ sq2,
                 float* __restrict__ cand)
{
  __shared__ float sA[kTile][kPitch];
  __shared__ float sB[kTile][kPitch];
  __shared__ float sDist[kTile * kTile];
  __shared__ float rVal[256];
  __shared__ int   rIdx[256];

  const int t    = threadIdx.x;
  const int wave = t >> 5;
  const int lane = t & 31;
  const int lo   = lane & 15;       // M (A) / N (B) lane index
  const int hi   = lane >> 4;       // selects K pair (A/B) and M half (C/D)
  const int scol  = wave & 3;       // subtile column 0..3
  const int srow0 = (wave >> 2) << 1; // first of two subtile rows
  const int p0 = blockIdx.y * kTile;
  const int q0 = blockIdx.x * kTile;

  v8f acc0 = {};
  v8f acc1 = {};

  for (int kc = 0; kc < kD; kc += kKC) {
    // Stage A/B chunk: 64 rows x 16 float4 each; 256 threads -> 4 float4 apiece.
    for (int idx = t; idx < kTile * (kKC / 4); idx += 256) {
      const int row = idx >> 4;
      const int c4  = (idx & 15) << 2;
      const float4 va = *reinterpret_cast<const float4*>(c1 + (size_t)(p0 + row) * kD + kc + c4);
      const float4 vb = *reinterpret_cast<const float4*>(c2 + (size_t)(q0 + row) * kD + kc + c4);
      *reinterpret_cast<float4*>(&sA[row][c4]) = va;
      *reinterpret_cast<float4*>(&sB[row][c4]) = vb;
      if (kc + kKC < kD) {
        __builtin_prefetch(c1 + (size_t)(p0 + row) * kD + kc + kKC + c4, 0, 0);
        __builtin_prefetch(c2 + (size_t)(q0 + row) * kD + kc + kKC + c4, 0, 0);
      }
    }
    __syncthreads();

    // 16 K-steps of 4; B fragment reused by both row subtiles.
    #pragma unroll
    for (int kk = 0; kk < kKC; kk += 4) {
      const int kb = kk + (hi << 1);
      const v2f bfrag  = *reinterpret_cast<const v2f*>(&sB[(scol << 4) + lo][kb]);
      const v2f afrag0 = *reinterpret_cast<const v2f*>(&sA[(srow0 << 4) + lo][kb]);
      const v2f afrag1 = *reinterpret_cast<const v2f*>(&sA[((srow0 + 1) << 4) + lo][kb]);
      acc0 = __builtin_amdgcn_wmma_f32_16x16x4_f32(false, afrag0, false, bfrag,
                                                   (short)0, acc0, false, false);
      acc1 = __builtin_amdgcn_wmma_f32_16x16x4_f32(false, afrag1, false, bfrag,
                                                   (short)0, acc1, false, false);
    }
    __syncthreads();
  }

  // sqrt(max(sq1 + sq2 - 2*dot, 0)) into the LDS tile.
  const int col = (scol << 4) + lo;
  const float sqc = sq2[q0 + col];
  #pragma unroll
  for (int r = 0; r < 8; ++r) {
    const int row0 = (srow0 << 4) + (hi << 3) + r;  // D layout: M = r + 8*hi
    const int row1 = row0 + 16;
    const float d0 = sq1[p0 + row0] + sqc - 2.0f * acc0[r];
    const float d1 = sq1[p0 + row1] + sqc - 2.0f * acc1[r];
    sDist[row0 * kTile + col] = sqrtf(fmaxf(d0, 0.0f));
    sDist[row1 * kTile + col] = sqrtf(fmaxf(d1, 0.0f));
  }
  __syncthreads();

  // Per-tile top-100 by repeated argmax extraction (descending order).
  float* candOut = cand + ((size_t)blockIdx.y * kGrid + blockIdx.x) * kTopK;
  for (int it = 0; it < kTopK; ++it) {
    float best = -1.0f;
    int bidx = 0;
    for (int i = t; i < kTile * kTile; i += 256) {
      const float v = sDist[i];
      if (v > best) { best = v; bidx = i; }
    }
    rVal[t] = best; rIdx[t] = bidx;
    __syncthreads();
    for (int s = 128; s > 0; s >>= 1) {
      if (t < s && rVal[t + s] > rVal[t]) { rVal[t] = rVal[t + s]; rIdx[t] = rIdx[t + s]; }
      __syncthreads();
    }
    if (t == 0) {
      candOut[it] = rVal[0];
      sDist[rIdx[0]] = -1.0f;
    }
    __syncthreads();
  }
}

// ---------------------------------------------------------------------------
// Kernel 3: global top-100 over 4096*100 candidates (L2-resident), mean.
// ---------------------------------------------------------------------------
__global__ void __launch_bounds__(256)
final_topk_kernel(float* __restrict__ cand, float* __restrict__ out)
{
  __shared__ float rVal[256];
  __shared__ int   rIdx[256];
  const int t = threadIdx.x;
  const int n = kBlocks * kTopK;
  float sum = 0.0f;

  for (int it = 0; it < kTopK; ++it) {
    float best = -1.0f;
    int bidx = 0;
    for (int i = t; i < n; i += 256) {
      const float v = cand[i];
      if (v > best) { best = v; bidx = i; }
    }
    rVal[t] = best; rIdx[t] = bidx;
    __syncthreads();
    for (int s = 128; s > 0; s >>= 1) {
      if (t < s && rVal[t + s] > rVal[t]) { rVal[t] = rVal[t + s]; rIdx[t] = rIdx[t + s]; }
      __syncthreads();
    }
    if (t == 0) {
      sum += rVal[0];
      cand[rIdx[0]] = -1.0f;   // visible to block after __syncthreads
    }
    __syncthreads();
  }
  if (t == 0) out[0] = sum / (float)kTopK;
}

// ---------------------------------------------------------------------------
extern "C" void kernel_launch(void* const* d_in, const int* in_sizes, int n_in,
                              void* d_out, int out_size, void* d_ws, size_t ws_size,
                              hipStream_t stream) {
  const float* fm   = (const float*)d_in[0];
  const int*   ids1 = (const int*)d_in[1];
  const int*   ids2 = (const int*)d_in[2];
  // d_in[3] = to_pick scalar; compile-time constant kTopK = 100 here.

  float* ws   = (float*)d_ws;
  float* c1   = ws;                                  // 4096*256
  float* c2   = c1 + (size_t)kPairs * kD;            // 4096*256
  float* sq1  = c2 + (size_t)kPairs * kD;            // 4096
  float* sq2  = sq1 + kPairs;                        // 4096
  float* cand = sq2 + kPairs;                        // 4096*100
  float* out  = (float*)d_out;

  gather_sq_kernel<<<kPairs, 256, 0, stream>>>(fm, ids1, ids2, c1, c2, sq1, sq2);
  dist_topk_kernel<<<dim3(kGrid, kGrid), 256, 0, stream>>>(c1, c2, sq1, sq2, cand);
  final_topk_kernel<<<1, 256, 0, stream>>>(cand, out);
}